// SimpleNet_TripleLayers_55800215109652
// MI455X (gfx1250) — compile-verified
//
#include <hip/hip_runtime.h>
#include <hip/hip_bf16.h>
#include <math.h>

// ---------------------------------------------------------------------------
// Fully fused conv pipeline for MI455X (gfx1250, wave32, WMMA).
//   y   = softplus(conv(x, w_motif, pad25))            [B,40,L]
//   s   = conv1x1(y, w_reduce); s = conv(s, w_fft, pad200)
//   g   = sigmoid(conv1x1(s, w_expand));  m = g * y
//   out = sigmoid(conv(m, w_effect, pad300))[:, :, 500:-500]
//
// One workgroup = 256 output positions of one batch row. All intermediates
// live in LDS (316KB of 320KB/WGP). Every conv is V_WMMA_F32_16X16X4_F32
// with K = 4 channels per step; channel-major LDS layouts make each WMMA
// operand fragment ONE conflict-free ds_load_b64, and clamped/duplicated
// weight rows keep every inner loop branch-free (no exec divergence).
// Phase 3 uses a shifted-A formulation: M rows = (4 out-ch) x (4 col-shifts)
// so one WMMA chain covers 64 output columns at full M utilization.
// ---------------------------------------------------------------------------

typedef float v2f __attribute__((ext_vector_type(2)));
typedef float v4f __attribute__((ext_vector_type(4)));
typedef float v8f __attribute__((ext_vector_type(8)));

#define LSEQ   65536
#define NBATCH 32
#define TB     256      // output positions per workgroup
#define NB1    79       // 16-col blocks covering y/s1 range (cols 0..1263)
#define NBG    54       // 16-col blocks covering gating range (864 cols)
#define NB3    14       // 64-col shifted blocks for fft conv (covers 896)
#define WXL    1314     // xs columns (1264 + 50 tap halo)
#define YSTR   52       // ys column stride: banks 4*(13l mod 16)+c0, no conflicts
#define S1COLS 1296     // 1264 + zeroed tail (shifted-A reads up to col 1295)
#define S2COLS 896      // 864 + overshoot cols (written, never read)
#define KT     32       // w_effect tap chunk staged in LDS
#define NCHUNK 19       // ceil(601/32)
#define OUTW   64536    // output slice [500, 65036)

// ---- LDS layout (floats) --------------------------------------------------
#define YS_OFF   0                        // ys   [1264][52]          65728
#define U0       65728                    // union region             15184
#define XS_OFF   (U0)                     //   ph0-1: xs  [1314][4]    5256
#define WML_OFF  (U0 + 5256)              //   ph0-1: wml [51][48][4]  9792
#define S1_OFF   (U0)                     //   ph2-3: s1  [1296][4]    5184
#define WFL_OFF  (U0 + 5184)              //   ph2-3: wfl [401][4][4]  6416
#define S2_OFF   (U0 + 5184 + 6416)       //   ph3-4: s2  [896][4]     3584
#define WEF_OFF  (U0)                     //   ph5:   weff[KT][2][40]  2560
#define LDS_FLOATS (65728 + 15184)        // 80912 floats = 323,648 B

__device__ __forceinline__ v8f wmma4(v2f a, v2f b, v8f c) {
#if defined(__HIP_DEVICE_COMPILE__)
  // D(16x16,f32) = A(16x4,f32) * B(4x16,f32) + C
  return __builtin_amdgcn_wmma_f32_16x16x4_f32(false, a, false, b, (short)0, c,
                                               false, false);
#else
  (void)a; (void)b;
  return c;
#endif
}

// Branch-free softplus/sigmoid on native v_exp_f32 / v_log_f32 / v_rcp_f32.
__device__ __forceinline__ float softplusf(float x) {
  return fmaxf(x, 0.f) + __logf(1.f + __expf(-fabsf(x)));
}
__device__ __forceinline__ float sigmoidf(float x) {
#if defined(__HIP_DEVICE_COMPILE__)
  return __builtin_amdgcn_rcpf(1.f + __expf(-x));
#else
  return 1.f / (1.f + expf(-x));
#endif
}
__device__ __forceinline__ int imin(int a, int b) { return a < b ? a : b; }
__device__ __forceinline__ int imax(int a, int b) { return a > b ? a : b; }

__global__ void __launch_bounds__(256, 1)
fused_pipeline_kernel(const float* __restrict__ x,
                      const float* __restrict__ wm, const float* __restrict__ bm,   // motif  [40][4][51]
                      const float* __restrict__ wr, const float* __restrict__ br,   // reduce [4][40]
                      const float* __restrict__ wf, const float* __restrict__ bf,   // fft    [4][4][401]
                      const float* __restrict__ we, const float* __restrict__ be,   // expand [40][4]
                      const float* __restrict__ wz, const float* __restrict__ bz,   // effect [2][40][601]
                      float* __restrict__ out) {
  extern __shared__ float sm[];
  float* ys   = sm + YS_OFF;
  float* xs   = sm + XS_OFF;
  float* wml  = sm + WML_OFF;
  float* s1   = sm + S1_OFF;
  float* wfl  = sm + WFL_OFF;
  float* s2   = sm + S2_OFF;
  float* weff = sm + WEF_OFF;

  const int tid  = threadIdx.x;
  const int wave = tid >> 5;
  const int lane = tid & 31;
  const int half = lane >> 4;       // lane half selects K-pair (c0, c0+1)
  const int n16  = lane & 15;       // N column / A row within half
  const int c0   = half * 2;
  const int bb   = blockIdx.y;
  const int l0   = blockIdx.x * TB;

  const float* xb = x + (size_t)bb * 4 * LSEQ;

  // ---- Phase 0: stage x tile (channel-major, branch-free zero pad) --------
  for (int c = 0; c < 4; ++c) {
    const float* xc = xb + (size_t)c * LSEQ;
    for (int j = tid; j < WXL; j += 256) {
      int p     = l0 - 525 + j;
      int psafe = imin(imax(p, 0), LSEQ - 1);
      float msk = (p >= 0 && p < LSEQ) ? 1.f : 0.f;
      xs[j * 4 + c] = xc[psafe] * msk;
    }
  }
  // stage motif weights, zero-padded rows 40..47: wml[k][m48][c]
  for (int i = tid; i < 51 * 48 * 4; i += 256) {
    int k = i / 192, r = i - k * 192;
    int m = r >> 2, c = r & 3;
    wml[i] = (m < 40) ? wm[(m * 4 + c) * 51 + k] : 0.f;
  }
  __syncthreads();

  // ---- Phase 1: motif conv (4->40, K=51) + softplus -> ys -----------------
  // ys col j holds position l0 - 500 + j, channel-major [col][52].
  v8f binit[3];
#pragma unroll
  for (int mt = 0; mt < 3; ++mt)
#pragma unroll
    for (int v = 0; v < 8; ++v)
      binit[mt][v] = bm[imin(mt * 16 + v + 8 * half, 39)];  // dup rows unused
  for (int nb = wave; nb < NB1; nb += 8) {
    const int j0 = nb * 16;
    v8f acc[3] = {binit[0], binit[1], binit[2]};
    for (int k = 0; k < 51; ++k) {
      v2f bv = *(const v2f*)&xs[(j0 + n16 + k) * 4 + c0];
#pragma unroll
      for (int mt = 0; mt < 3; ++mt) {
        v2f av  = *(const v2f*)&wml[(k * 48 + mt * 16 + n16) * 4 + c0];
        acc[mt] = wmma4(av, bv, acc[mt]);
      }
    }
    int col  = j0 + n16;
    int pos  = l0 - 500 + col;
    float zm = (pos >= 0 && pos < LSEQ) ? 1.f : 0.f;   // reference zero-pads y
    float* yrow = &ys[col * YSTR];
#pragma unroll
    for (int mt = 0; mt < 3; ++mt) {
      v4f lo, hi;
#pragma unroll
      for (int v = 0; v < 4; ++v) lo[v] = softplusf(acc[mt][v]) * zm;
#pragma unroll
      for (int v = 0; v < 4; ++v) hi[v] = softplusf(acc[mt][v + 4]) * zm;
      *(v4f*)&yrow[mt * 16 + 8 * half]     = lo;       // ch block contiguous
      *(v4f*)&yrow[mt * 16 + 8 * half + 4] = hi;
    }
  }
  __syncthreads();

  // ---- stage fft weights (overlays dead wml/xs tail): wfl[k][o][c] --------
  for (int i = tid; i < 401 * 16; i += 256) {
    int k = i >> 4, r = i & 15;
    int m = r >> 2, c = r & 3;
    wfl[i] = wf[(m * 4 + c) * 401 + k];
  }
  // zero s1 tail cols [1264,1296) so masked shifted-A products stay finite
  if (tid < (S1COLS - 1264) * 4) s1[1264 * 4 + tid] = 0.f;

  // ---- Phase 2: reduce 1x1 (40->4), K = 40 in 10 groups of 4 --------------
  const int m4 = n16 & 3;   // lanes 4..15 duplicate rows 0..3 (D rows unused)
  {
    v8f binit2;
#pragma unroll
    for (int v = 0; v < 8; ++v) binit2[v] = br[imin(v + 8 * half, 3)];
    for (int nb = wave; nb < NB1; nb += 8) {
      const int j0 = nb * 16;
      v8f acc = binit2;
#pragma unroll
      for (int kk = 0; kk < 10; ++kk) {
        int ch = 4 * kk + c0;
        v2f av = *(const v2f*)&wr[m4 * 40 + ch];
        v2f bv = *(const v2f*)&ys[(j0 + n16) * YSTR + ch];
        acc = wmma4(av, bv, acc);
      }
      int col  = j0 + n16;
      int pos  = l0 - 500 + col;
      float zm = (pos >= 0 && pos < LSEQ) ? 1.f : 0.f;
      if (half == 0) {
        v4f st = {acc[0] * zm, acc[1] * zm, acc[2] * zm, acc[3] * zm};
        *(v4f*)&s1[col * 4] = st;
      }
    }
  }
  __syncthreads();

  // ---- Phase 3: fft conv (4->4, K=401) -> s2, shifted-A full-M form -------
  // M row m = (js = m>>2, o = m&3): covers s2 cols j0 + 16*js + n16 for one
  // 64-col block. K' = 449 steps; A[(o,js)][(c,k')] = wf[o][c][k'-16*js],
  // zero-masked outside [0,401). s2 col g holds position l0 - 300 + g.
  {
    const int jsh = (n16 >> 2) * 16;            // lane's A-row column shift
    v8f binit3;
#pragma unroll
    for (int v = 0; v < 8; ++v) binit3[v] = bf[(v + 8 * half) & 3];
    for (int nb = wave; nb < NB3; nb += 8) {
      const int j0 = nb * 64;
      v8f acc = binit3;
      for (int kp = 0; kp < 449; ++kp) {
        int kk    = kp - jsh;
        int kks   = imin(imax(kk, 0), 400);
        float vmk = ((unsigned)kk <= 400u) ? 1.f : 0.f;
        v2f av = *(const v2f*)&wfl[(kks * 4 + m4) * 4 + c0];
        av[0] *= vmk;
        av[1] *= vmk;
        v2f bv = *(const v2f*)&s1[(j0 + n16 + kp) * 4 + c0];
        acc = wmma4(av, bv, acc);
      }
#pragma unroll
      for (int v = 0; v < 8; ++v) {
        int m = v + 8 * half;                   // D row -> (js, o)
        s2[(j0 + (m >> 2) * 16 + n16) * 4 + (m & 3)] = acc[v];
      }
    }
  }
  __syncthreads();

  // ---- Phase 4: expand 1x1 (4->40) + sigmoid gate, ys *= gate (in place) --
  // Gating range = ys cols [200, 1064). Dup rows 40..47 gated too (unused).
  {
    v8f binit4[3];
#pragma unroll
    for (int mt = 0; mt < 3; ++mt)
#pragma unroll
      for (int v = 0; v < 8; ++v)
        binit4[mt][v] = be[imin(mt * 16 + v + 8 * half, 39)];
    for (int nb = wave; nb < NBG; nb += 8) {
      const int j0 = nb * 16;
      v2f bv = *(const v2f*)&s2[(j0 + n16) * 4 + c0];
      float* yrow = &ys[(200 + j0 + n16) * YSTR];
#pragma unroll
      for (int mt = 0; mt < 3; ++mt) {
        v8f acc = binit4[mt];
        v2f av  = *(const v2f*)&we[imin(mt * 16 + n16, 39) * 4 + c0];
        acc = wmma4(av, bv, acc);
        int base = mt * 16 + 8 * half;
        v4f y0 = *(v4f*)&yrow[base];
        v4f y1 = *(v4f*)&yrow[base + 4];
#pragma unroll
        for (int v = 0; v < 4; ++v) y0[v] *= sigmoidf(acc[v]);
#pragma unroll
        for (int v = 0; v < 4; ++v) y1[v] *= sigmoidf(acc[v + 4]);
        *(v4f*)&yrow[base]     = y0;
        *(v4f*)&yrow[base + 4] = y1;
      }
    }
  }
  __syncthreads();

  // ---- Phase 5: effect conv (40->2, K=601) + sigmoid -> out ---------------
  // Each wave owns 2 adjacent 16-col output blocks. w_effect (192KB) streams
  // through LDS in KT-tap chunks: weff[kp][m][c], rows 2..15 dup row 0/1.
  const int j0a = (2 * wave) * 16;
  const int j0b = j0a + 16;
  const int mA  = n16 & 1;
  v8f acc0, acc1;
#pragma unroll
  for (int v = 0; v < 8; ++v) {
    float bi = bz[imin(v + 8 * half, 1)];
    acc0[v] = bi;
    acc1[v] = bi;
  }
  for (int ck = 0; ck < NCHUNK; ++ck) {
    __syncthreads();                            // prev chunk fully consumed
    for (int i = tid; i < KT * 80; i += 256) {
      int kp = i / 80, r = i - kp * 80;
      int m  = r / 40, c = r - m * 40;
      int k  = ck * KT + kp;
      float msk = (k < 601) ? 1.f : 0.f;
      weff[i] = wz[(m * 40 + c) * 601 + imin(k, 600)] * msk;
    }
    __syncthreads();                            // chunk visible to all waves
    int kend = 601 - ck * KT;
    if (kend > KT) kend = KT;
    for (int kp = 0; kp < kend; ++kp) {
      int off = 200 + n16 + ck * KT + kp;       // motifact col = out col+k+200
      v2f A[10];
#pragma unroll
      for (int kk = 0; kk < 10; ++kk)
        A[kk] = *(const v2f*)&weff[(kp * 2 + mA) * 40 + 4 * kk + c0];
#pragma unroll
      for (int kk = 0; kk < 10; ++kk) {
        int ch = 4 * kk + c0;
        v2f b0 = *(const v2f*)&ys[(j0a + off) * YSTR + ch];
        acc0   = wmma4(A[kk], b0, acc0);
        v2f b1 = *(const v2f*)&ys[(j0b + off) * YSTR + ch];
        acc1   = wmma4(A[kk], b1, acc1);
      }
    }
  }
  if (half == 0) {
#pragma unroll
    for (int v = 0; v < 2; ++v) {               // D rows 0,1 = output channels
      int pa = l0 + j0a + n16;
      if (pa >= 500 && pa < 65036)
        out[((size_t)bb * 2 + v) * OUTW + (pa - 500)] = sigmoidf(acc0[v]);
      int pb = l0 + j0b + n16;
      if (pb >= 500 && pb < 65036)
        out[((size_t)bb * 2 + v) * OUTW + (pb - 500)] = sigmoidf(acc1[v]);
    }
  }
}

extern "C" void kernel_launch(void* const* d_in, const int* in_sizes, int n_in,
                              void* d_out, int out_size, void* d_ws, size_t ws_size,
                              hipStream_t stream) {
  (void)in_sizes; (void)n_in; (void)out_size; (void)d_ws; (void)ws_size;
  const float* x  = (const float*)d_in[0];
  const float* wm = (const float*)d_in[1];
  const float* bm = (const float*)d_in[2];
  const float* wr = (const float*)d_in[3];
  const float* br = (const float*)d_in[4];
  const float* wf = (const float*)d_in[5];
  const float* bf = (const float*)d_in[6];
  const float* we = (const float*)d_in[7];
  const float* be = (const float*)d_in[8];
  const float* wz = (const float*)d_in[9];
  const float* bz = (const float*)d_in[10];
  float* out = (float*)d_out;

  size_t shmem = (size_t)LDS_FLOATS * sizeof(float);
  // Opt in to >64KB dynamic LDS (gfx1250 allows up to 320KB per workgroup).
  hipFuncSetAttribute(reinterpret_cast<const void*>(fused_pipeline_kernel),
                      hipFuncAttributeMaxDynamicSharedMemorySize, (int)shmem);

  dim3 grid(LSEQ / TB, NBATCH);
  fused_pipeline_kernel<<<grid, 256, shmem, stream>>>(x, wm, bm, wr, br, wf, bf,
                                                      we, be, wz, bz, out);
}